// ROIRotate_12240656794020
// MI455X (gfx1250) — compile-verified
//
#include <hip/hip_runtime.h>
#include <stdint.h>

#define M_BOXES 1024
#define N_BATCH 16
#define C_CH 32
#define HF_DIM 128
#define WF_DIM 128
#define HEIGHT 8
#define W_MAX 96
#define NWAVES (M_BOXES / 32)   // 32 waves in the rank block

struct BoxParam {
  float p1x, p1y, ux, uy, vx, vy;
  int width, n;
};

// ---------------- Kernel 1: per-box geometry ----------------
__global__ void box_param_kernel(const float* __restrict__ boxes,
                                 const int* __restrict__ mapping,
                                 BoxParam* __restrict__ params,
                                 int* __restrict__ widths) {
  int i = blockIdx.x * blockDim.x + threadIdx.x;
  if (i >= M_BOXES) return;
  const float* b = boxes + i * 8;
  float p1x = b[0] * 0.25f, p1y = b[1] * 0.25f;
  float p2x = b[2] * 0.25f, p2y = b[3] * 0.25f;
  float p4x = b[6] * 0.25f, p4y = b[7] * 0.25f;
  float dwx = p2x - p1x, dwy = p2y - p1y;
  float dhx = p4x - p1x, dhy = p4y - p1y;
  float ew = sqrtf(dwx * dwx + dwy * dwy);
  float eh = sqrtf(dhx * dhx + dhy * dhy);
  float bw = fmaxf(ew, eh), bh = fminf(ew, eh);
  float wfl = ceilf((float)HEIGHT * bw / (bh + 1e-6f));
  wfl = fminf(fmaxf(wfl, 1.0f), (float)W_MAX);
  BoxParam p;
  p.p1x = p1x; p.p1y = p1y;
  p.ux = dwx / wfl;           p.uy = dwy / wfl;
  p.vx = dhx / (float)HEIGHT; p.vy = dhy / (float)HEIGHT;
  p.width = (int)wfl;
  p.n = mapping[i];
  params[i] = p;
  widths[i] = p.width;
}

// ---------------- Kernel 2: stable counting sort by -width ----------------
// Reproduces jnp.argsort(-width) (stable): key k = W_MAX - width ascending,
// ties broken by original index. One block of 1024 threads (32 waves).
// rank = start[k] + (#equal-width boxes in earlier waves) + (#equal-width
// boxes among lower-index lanes in my wave).
__global__ __launch_bounds__(M_BOXES)
void rank_kernel(const int* __restrict__ widths,
                 int* __restrict__ order_int,
                 float* __restrict__ out_tail) {
  __shared__ int s_w[M_BOXES];
  __shared__ int s_cnt[W_MAX + 1];
  __shared__ int s_start[W_MAX + 1];
  __shared__ int s_whist[NWAVES][W_MAX + 1];   // per-wave bin counts
  const int tid = threadIdx.x;
  const int wave = tid >> 5;
  const int waveBase = wave << 5;

  // CDNA5 async global->LDS copy of the widths array (ASYNCcnt-tracked)
  {
    uint32_t lds_off = (uint32_t)(uintptr_t)(&s_w[tid]);
    uint64_t gaddr = (uint64_t)(uintptr_t)(widths + tid);
    asm volatile("global_load_async_to_lds_b32 %0, %1, off"
                 :
                 : "v"(lds_off), "v"(gaddr)
                 : "memory");
  }
  // Zero histograms while the async copy is in flight.
  if (tid <= W_MAX) s_cnt[tid] = 0;
  for (int e = tid; e < NWAVES * (W_MAX + 1); e += M_BOXES)
    ((int*)s_whist)[e] = 0;
  asm volatile("s_wait_asynccnt 0" ::: "memory");
  __syncthreads();

  const int w = s_w[tid];
  const int k = W_MAX - w;
  atomicAdd(&s_cnt[k], 1);
  atomicAdd(&s_whist[wave][k], 1);
  __syncthreads();

  if (tid == 0) {
    int acc = 0;
    for (int j = 0; j <= W_MAX; ++j) { s_start[j] = acc; acc += s_cnt[j]; }
  }
  __syncthreads();

  // Earlier waves' contribution (<= 31 iterations)
  int tie = 0;
  for (int wv = 0; wv < wave; ++wv) tie += s_whist[wv][k];
  // Lower-index lanes in my wave (<= 31 iterations, broadcast LDS reads)
  for (int j = waveBase; j < tid; ++j) tie += (s_w[j] == w) ? 1 : 0;
  const int rank = s_start[k] + tie;

  order_int[rank] = tid;                   // order[rank] = original index
  out_tail[rank] = (float)w;               // width[order]
  out_tail[M_BOXES + rank] = (float)tid;   // order (as float in flat f32 d_out)
}

// ---------------- Kernel 3: gather + bilinear, write into sorted slot ----------------
// Block = (32, 8): lane = channel (wave32 == C), one wave per grid row, one
// block per output slot. Wave stores one coalesced 128B line per sample via
// non-temporal stores (output is write-once: keep L2 for the L2-resident
// feature-map gathers). global_prefetch_b8 pulls tap lines 8 samples ahead.
__global__ __launch_bounds__(C_CH * HEIGHT)
void roi_rotate_kernel(const float* __restrict__ fm,
                       const BoxParam* __restrict__ params,
                       const int* __restrict__ order_int,
                       float* __restrict__ out) {
  const int slot = blockIdx.x;
  const int c = threadIdx.x;   // channel = lane
  const int y = threadIdx.y;   // grid row handled by this wave
  const int box = order_int[slot];
  const BoxParam p = params[box];

  const float sxb = p.p1x + (float)y * p.vx;
  const float syb = p.p1y + (float)y * p.vy;
  const float* __restrict__ fmc =
      fm + (size_t)(p.n * C_CH + c) * (size_t)(HF_DIM * WF_DIM);
  float* __restrict__ orow =
      out + ((size_t)slot * HEIGHT + y) * (size_t)(W_MAX * C_CH) + c;
  const int width = p.width;   // uniform across the block

  for (int x = 0; x < width; ++x) {
    float sx = sxb + (float)x * p.ux;
    float sy = syb + (float)x * p.uy;
    float x0f = floorf(sx), y0f = floorf(sy);
    float wx = sx - x0f, wy = sy - y0f;
    int x0 = (int)x0f, y0 = (int)y0f;
    int x1 = x0 + 1, y1 = y0 + 1;
    int x0c = min(max(x0, 0), WF_DIM - 1);
    int x1c = min(max(x1, 0), WF_DIM - 1);
    int y0c = min(max(y0, 0), HF_DIM - 1);
    int y1c = min(max(y1, 0), HF_DIM - 1);

    float g00 = fmc[y0c * WF_DIM + x0c];
    float g01 = fmc[y0c * WF_DIM + x1c];
    float g10 = fmc[y1c * WF_DIM + x0c];
    float g11 = fmc[y1c * WF_DIM + x1c];

    bool vx0 = (x0 >= 0) && (x0 < WF_DIM);
    bool vx1 = (x1 >= 0) && (x1 < WF_DIM);
    bool vy0 = (y0 >= 0) && (y0 < HF_DIM);
    bool vy1 = (y1 >= 0) && (y1 < HF_DIM);
    float v00 = (vy0 && vx0) ? g00 : 0.0f;
    float v01 = (vy0 && vx1) ? g01 : 0.0f;
    float v10 = (vy1 && vx0) ? g10 : 0.0f;
    float v11 = (vy1 && vx1) ? g11 : 0.0f;

    float omwx = 1.0f - wx, omwy = 1.0f - wy;
    float r = v00 * (omwy * omwx) + v01 * (omwy * wx)
            + v10 * (wy * omwx)  + v11 * (wy * wx);

    if ((x & 7) == 0) {
      int pxc = min(max((int)(sx + 8.0f * p.ux), 0), WF_DIM - 1);
      int pyc = min(max((int)(sy + 8.0f * p.uy), 0), HF_DIM - 1);
      __builtin_prefetch(&fmc[pyc * WF_DIM + pxc], 0, 1);
    }

    __builtin_nontemporal_store(r, &orow[(size_t)x * C_CH]);
  }
  // Zero-fill masked columns (x >= width), also write-once -> non-temporal.
  for (int x = width; x < W_MAX; ++x) {
    __builtin_nontemporal_store(0.0f, &orow[(size_t)x * C_CH]);
  }
}

// ---------------- Launch ----------------
extern "C" void kernel_launch(void* const* d_in, const int* in_sizes, int n_in,
                              void* d_out, int out_size, void* d_ws, size_t ws_size,
                              hipStream_t stream) {
  const float* fm = (const float*)d_in[0];     // (16,32,128,128) f32
  const float* boxes = (const float*)d_in[1];  // (1024,8) f32
  const int* mapping = (const int*)d_in[2];    // (1024,) i32
  float* out = (float*)d_out;

  char* ws = (char*)d_ws;
  BoxParam* params = (BoxParam*)ws;                                   // 32 KB
  int* widths = (int*)(ws + sizeof(BoxParam) * M_BOXES);              // 4 KB
  int* order_int = (int*)(ws + sizeof(BoxParam) * M_BOXES
                             + sizeof(int) * M_BOXES);                // 4 KB

  box_param_kernel<<<(M_BOXES + 255) / 256, 256, 0, stream>>>(
      boxes, mapping, params, widths);

  float* out_tail = out + (size_t)M_BOXES * HEIGHT * W_MAX * C_CH;
  rank_kernel<<<1, M_BOXES, 0, stream>>>(widths, order_int, out_tail);

  dim3 blk(C_CH, HEIGHT);
  roi_rotate_kernel<<<M_BOXES, blk, 0, stream>>>(fm, params, order_int, out);
}